// SparseCommunicationNetwork_38001870635679
// MI455X (gfx1250) — compile-verified
//
#include <hip/hip_runtime.h>
#include <math.h>

// SparseCommunicationNetwork for MI455X (gfx1250, wave32).
// Memory-bound (~95 MB weight traffic ≈ 4 us @ 23.3 TB/s). fp32 throughout to
// preserve the discrete top-K decision. WMMA f32 16x16x4 used for the two
// [N,64]@vec kernels; exact JAX top_k ordering via stable 4-bit LSD radix sort
// on ~bits(sigmoid) with index payload.

#define OBS  131072
#define TOPK 39321

typedef __attribute__((ext_vector_type(2))) float v2f;
typedef __attribute__((ext_vector_type(8))) float v8f;

// ---------------------------------------------------------------- zero
__global__ void k_zero_f32(float* __restrict__ p, int n) {
  int i = blockIdx.x * blockDim.x + threadIdx.x;
  if (i < n) p[i] = 0.f;
}

// ------------------------------------------------ wide row-dot partials
// W[rows, n] @ x, split into `chunks` column chunks per row.
// grid.x = rows*chunks, block = 256. Coalesced float4 streaming of W.
__global__ __launch_bounds__(256) void k_partial_dot(
    const float* __restrict__ W, const float* __restrict__ x,
    float* __restrict__ partials, int chunks, int n) {
  int row = blockIdx.x / chunks, chunk = blockIdx.x % chunks;
  int clen = n / chunks;
  const float4* wr = (const float4*)(W + (size_t)row * n + (size_t)chunk * clen);
  const float4* xr = (const float4*)(x + (size_t)chunk * clen);
  float s = 0.f;
  int nq = clen >> 2;
  for (int i = threadIdx.x; i < nq; i += 256) {
    float4 a = wr[i], b = xr[i];
    s += a.x * b.x + a.y * b.y + a.z * b.z + a.w * b.w;
  }
  __shared__ float red[256];
  red[threadIdx.x] = s;
  __syncthreads();
  for (int o = 128; o > 0; o >>= 1) {
    if (threadIdx.x < o) red[threadIdx.x] += red[threadIdx.x + o];
    __syncthreads();
  }
  if (threadIdx.x == 0) partials[blockIdx.x] = red[0];
}

__global__ void k_finalize_relu(const float* __restrict__ partials, int chunks,
                                const float* __restrict__ bias,
                                float* __restrict__ out, int rows) {
  int r = threadIdx.x;
  if (r < rows) {
    float s = bias[r];
    for (int c = 0; c < chunks; ++c) s += partials[r * chunks + c];
    out[r] = s > 0.f ? s : 0.f;
  }
}

// ------------------------------------------- WMMA [N,64] @ vec kernels
// One wave computes a 16-row tile of y = W[rows,64] @ v via 16 chained
// V_WMMA_F32_16X16X4_F32. A = 16x4 tile of W (fp32 A layout per ISA:
// lanes 0-15 hold K=0,1; lanes 16-31 hold K=2,3). B has the vector in
// column N=0 only (lanes 0 and 16), rest zero. Column 0 of D is read
// back from lanes 0 (M=0..7) and 16 (M=8..15).
//
// B fragments are built branch-free (uniform LDS load + mask multiply)
// and hoisted before the loop; A fragments are batch-loaded so the
// accumulate chain runs as a clean burst of global_load_b64 then wmma.
__device__ __forceinline__ v8f wmma_rowdot64(const float* __restrict__ wrow,
                                             const float* __restrict__ sv,
                                             int lane) {
  const int koff = (lane >> 4) << 1;                  // 0 or 2
  const float m = ((lane & 15) == 0) ? 1.f : 0.f;     // column-0 mask
  v2f a[16], b[16];
#pragma unroll
  for (int s = 0; s < 16; ++s) {
    int k0 = s * 4 + koff;
    // uniform LDS read by all lanes, masked into column 0 of B
    b[s].x = sv[k0] * m;
    b[s].y = sv[k0 + 1] * m;
    // A fragment: 2 consecutive fp32 of this lane's W row
    a[s].x = wrow[k0];
    a[s].y = wrow[k0 + 1];
  }
  v8f c = {};
#pragma unroll
  for (int s = 0; s < 16; ++s) {
    c = __builtin_amdgcn_wmma_f32_16x16x4_f32(false, a[s], false, b[s],
                                              (short)0, c, false, false);
  }
  return c;
}

// importance = sigmoid(w_imp2 @ h + b); emit sort keys ~bits(sig) + index.
__global__ __launch_bounds__(256) void k_imp2_wmma(
    const float* __restrict__ W, const float* __restrict__ v,
    const float* __restrict__ bias, unsigned* __restrict__ keyOut,
    unsigned* __restrict__ payOut) {
  __shared__ float sv[64];
  if (threadIdx.x < 64) sv[threadIdx.x] = v[threadIdx.x];
  __syncthreads();
  int lane = threadIdx.x & 31;
  int tile = blockIdx.x * 8 + (threadIdx.x >> 5);
  const float* wr = W + (size_t)(tile * 16 + (lane & 15)) * 64;
  v8f c = wmma_rowdot64(wr, sv, lane);
  if ((lane & 15) == 0) {
    int mbase = (lane >> 4) * 8;
#pragma unroll
    for (int r = 0; r < 8; ++r) {
      int row = tile * 16 + mbase + r;
      float z = c[r] + bias[row];
      float sig = 1.f / (1.f + expf(-z));
      keyOut[row] = ~__float_as_uint(sig);  // ascending key == descending sig
      payOut[row] = (unsigned)row;
    }
  }
}

// out[idx[j]] = w_dec2[j,:] @ d + b_dec2[j]  for j < K (idx = sorted payload)
__global__ __launch_bounds__(256) void k_dec2_wmma(
    const float* __restrict__ W, const float* __restrict__ v,
    const float* __restrict__ bias, const unsigned* __restrict__ pay,
    float* __restrict__ out, int K) {
  __shared__ float sv[64];
  if (threadIdx.x < 64) sv[threadIdx.x] = v[threadIdx.x];
  __syncthreads();
  int lane = threadIdx.x & 31;
  int tile = blockIdx.x * 8 + (threadIdx.x >> 5);
  // rows < 2464*16 = 39424 << 131072, loads always in-bounds
  const float* wr = W + (size_t)(tile * 16 + (lane & 15)) * 64;
  v8f c = wmma_rowdot64(wr, sv, lane);
  if ((lane & 15) == 0) {
    int mbase = (lane >> 4) * 8;
#pragma unroll
    for (int r = 0; r < 8; ++r) {
      int j = tile * 16 + mbase + r;
      if (j < K) out[pay[j]] = c[r] + bias[j];
    }
  }
}

// ---------------------------------------------- stable LSD radix sort
// 8 passes x 4 bits, single workgroup of 512 threads, 256 elems/thread.
// Threads own contiguous index chunks processed sequentially -> stable,
// reproducing jax.lax.top_k tie order (lower index first on equal values).
// Total traffic ~2 MB/pass, L2-resident; small fraction of the 95 MB
// weight streaming, so a single-WGP sorter is acceptable.
#define ST 512
#define EPT 256
__global__ __launch_bounds__(ST) void k_radix_sort(unsigned* keyA, unsigned* payA,
                                                   unsigned* keyB, unsigned* payB) {
  __shared__ unsigned cnt[ST * 16];  // 32 KB
  __shared__ unsigned aux[ST];
  const int t = threadIdx.x;
  for (int pass = 0; pass < 8; ++pass) {
    int shift = pass * 4;
    unsigned* sk = (pass & 1) ? keyB : keyA;
    unsigned* sp = (pass & 1) ? payB : payA;
    unsigned* dk = (pass & 1) ? keyA : keyB;
    unsigned* dp = (pass & 1) ? payA : payB;
    unsigned lc[16];
#pragma unroll
    for (int d = 0; d < 16; ++d) lc[d] = 0;
    int base = t * EPT;
    for (int e = 0; e < EPT; ++e) lc[(sk[base + e] >> shift) & 15]++;
#pragma unroll
    for (int d = 0; d < 16; ++d) cnt[d * ST + t] = lc[d];  // digit-major
    __syncthreads();
    // block-wide exclusive scan over the 8192 flattened counters
    unsigned s = 0;
#pragma unroll
    for (int i = 0; i < 16; ++i) s += cnt[t * 16 + i];
    aux[t] = s;
    __syncthreads();
    for (int off = 1; off < ST; off <<= 1) {
      unsigned vv = (t >= off) ? aux[t - off] : 0u;
      __syncthreads();
      aux[t] += vv;
      __syncthreads();
    }
    unsigned run = (t == 0) ? 0u : aux[t - 1];
#pragma unroll
    for (int i = 0; i < 16; ++i) {
      unsigned cc = cnt[t * 16 + i];
      cnt[t * 16 + i] = run;
      run += cc;
    }
    __syncthreads();
#pragma unroll
    for (int d = 0; d < 16; ++d) lc[d] = cnt[d * ST + t];
    for (int e = 0; e < EPT; ++e) {
      unsigned k = sk[base + e];
      unsigned d = (k >> shift) & 15;
      unsigned dst = lc[d]++;
      dk[dst] = k;
      dp[dst] = sp[base + e];
    }
    __syncthreads();
  }
  // 8 passes (even) -> sorted result ends back in keyA/payA
}

// ----------------------------------------------------------- scatters
__global__ void k_scatter_x(const unsigned* __restrict__ pay,
                            const float* __restrict__ x,
                            float* __restrict__ sx, int K) {
  int j = blockIdx.x * blockDim.x + threadIdx.x;
  if (j < K) { unsigned i = pay[j]; sx[i] = x[i]; }
}

__global__ void k_dec1(const float* __restrict__ w, const float* __restrict__ b,
                       const float* __restrict__ enc, float* __restrict__ d) {
  int r = threadIdx.x;
  if (r < 64) {
    float s = b[r];
    for (int k = 0; k < 32; ++k) s += w[r * 32 + k] * enc[k];
    d[r] = s > 0.f ? s : 0.f;
  }
}

// ------------------------------------------------------------- launch
extern "C" void kernel_launch(void* const* d_in, const int* in_sizes, int n_in,
                              void* d_out, int out_size, void* d_ws, size_t ws_size,
                              hipStream_t stream) {
  const float* x      = (const float*)d_in[0];
  const float* w_imp1 = (const float*)d_in[1];
  const float* b_imp1 = (const float*)d_in[2];
  const float* w_imp2 = (const float*)d_in[3];
  const float* b_imp2 = (const float*)d_in[4];
  const float* w_enc  = (const float*)d_in[5];
  const float* b_enc  = (const float*)d_in[6];
  const float* w_dec1 = (const float*)d_in[7];
  const float* b_dec1 = (const float*)d_in[8];
  const float* w_dec2 = (const float*)d_in[9];
  const float* b_dec2 = (const float*)d_in[10];
  float* out = (float*)d_out;

  // workspace layout (~2.6 MB total)
  char* ws = (char*)d_ws;
  unsigned* keyA    = (unsigned*)(ws + 0);        // 512 KB
  unsigned* payA    = (unsigned*)(ws + 524288);   // 512 KB
  unsigned* keyB    = (unsigned*)(ws + 1048576);  // 512 KB
  unsigned* payB    = (unsigned*)(ws + 1572864);  // 512 KB
  float*    sparse_x= (float*)(ws + 2097152);     // 512 KB
  float*    partials= (float*)(ws + 2621440);     // 8 KB (reused)
  float*    hvec    = (float*)(ws + 2631680);
  float*    enc     = (float*)(ws + 2632192);
  float*    dvec    = (float*)(ws + 2632704);

  // h = relu(w_imp1 @ x + b)
  k_partial_dot<<<64 * 32, 256, 0, stream>>>(w_imp1, x, partials, 32, OBS);
  k_finalize_relu<<<1, 64, 0, stream>>>(partials, 32, b_imp1, hvec, 64);

  // importance keys (sigmoid fused, WMMA): 8192 tiles, 8 waves/block
  k_imp2_wmma<<<1024, 256, 0, stream>>>(w_imp2, hvec, b_imp2, keyA, payA);

  // exact stable top-K ordering
  k_radix_sort<<<1, ST, 0, stream>>>(keyA, payA, keyB, payB);

  // sparse_x and output zeroing + gather mask
  k_zero_f32<<<OBS / 256, 256, 0, stream>>>(sparse_x, OBS);
  k_zero_f32<<<OBS / 256, 256, 0, stream>>>(out, OBS);
  k_scatter_x<<<(TOPK + 255) / 256, 256, 0, stream>>>(payA, x, sparse_x, TOPK);

  // encoded = relu(w_enc @ sparse_x + b)
  k_partial_dot<<<32 * 32, 256, 0, stream>>>(w_enc, sparse_x, partials, 32, OBS);
  k_finalize_relu<<<1, 32, 0, stream>>>(partials, 32, b_enc, enc, 32);

  // d = relu(w_dec1 @ encoded + b)
  k_dec1<<<1, 64, 0, stream>>>(w_dec1, b_dec1, enc, dvec);

  // out[idx[j]] = w_dec2[j,:] @ d + b_dec2[j] (WMMA, scattered store)
  int tiles = (TOPK + 15) / 16;        // 2458
  int blocks = (tiles + 7) / 8;        // 308
  k_dec2_wmma<<<blocks, 256, 0, stream>>>(w_dec2, dvec, b_dec2, payA, out, TOPK);
}